// LCA_44873818309115
// MI455X (gfx1250) — compile-verified
//
#include <hip/hip_runtime.h>
#include <math.h>

// ---------------------------------------------------------------------------
// LCA sparse coding on MI455X (gfx1250), f32 WMMA path.
// B=256 samples, N=1024 signal dim, K=4096 dictionary atoms, 30 iterations.
// Dictionary operand is stored row-pair-interleaved ("packed") so each lane's
// B fragment is one contiguous float2 -> single global_load_b64 per fragment.
// ---------------------------------------------------------------------------

typedef __attribute__((ext_vector_type(2))) float v2f;
typedef __attribute__((ext_vector_type(8))) float v8f;

#define LCA_B 256
#define LCA_N 1024
#define LCA_K 4096
#define LCA_BK (LCA_B * LCA_K)   // 1048576
#define LCA_BN (LCA_B * LCA_N)   // 262144
#define LCA_LAMBD 0.1f
#define LCA_TOL 1e-6f
#define LCA_EPS 1e-12f
#define LCA_ITERS 30

// ---------------------------------------------------------------------------
// Generic batched WMMA GEMM: C[bat] = A[bat] * Bpk (+ addC).
// A row-major [M,K]; Bpk is the packed form of row-major B [K,N]:
//   Bpk[ ((k>>1)*ldb + n)*2 + (k&1) ] = B[k][n]
// Each wave computes a 32x32 tile as 2x2 grid of 16x16 f32 WMMA accumulators.
// blockDim.x must be 256 (8 waves); gridDim.x * 8 == (M/32)*(N/32); EXEC all 1s.
// ---------------------------------------------------------------------------
__global__ void lca_wmma_gemm32(const float* __restrict__ A,
                                const float* __restrict__ Bpk,
                                float* __restrict__ C,
                                int N, int K, int lda, int ldb, int ldc,
                                long batchStrideA, long batchStrideC,
                                float addC)
{
    A += (size_t)blockIdx.y * (size_t)batchStrideA;
    C += (size_t)blockIdx.y * (size_t)batchStrideC;

    const int wave = threadIdx.x >> 5;
    const int lane = threadIdx.x & 31;
    const int quad = lane >> 4;   // 0: lanes 0-15, 1: lanes 16-31
    const int mn   = lane & 15;   // M index for A frags, N index for B/C frags

    const int tilesN = N >> 5;
    const int tile = blockIdx.x * 8 + wave;
    const int m0 = (tile / tilesN) << 5;
    const int n0 = (tile % tilesN) << 5;

    v8f acc00 = {}, acc01 = {}, acc10 = {}, acc11 = {};

    // A 16x4 fragment (ISA 7.12.2): lane m = lane%16,
    //   component v holds K = k0 + v + 2*quad  -> contiguous float2 per lane.
    const float* a0p = A + (size_t)(m0 + mn) * lda + 2 * quad;
    const float* a1p = a0p + (size_t)16 * lda;
    // B 4x16 fragment: component v holds row K = k0 + v + 2*quad, col n0+mn.
    // Packed: pair index p = k0/2 + quad, contiguous float2 at (p*ldb + col)*2.
    const float* b0p = Bpk + (((size_t)quad * ldb) + n0 + mn) * 2;
    const float* b1p = b0p + 32;                 // +16 columns * 2
    const size_t bstride = (size_t)ldb * 2;      // floats per pair-row

    for (int k0 = 0; k0 < K; k0 += 4) {
        v2f af0 = *(const v2f*)(a0p + k0);
        v2f af1 = *(const v2f*)(a1p + k0);
        const size_t boff = (size_t)(k0 >> 1) * bstride;
        v2f bf0 = *(const v2f*)(b0p + boff);
        v2f bf1 = *(const v2f*)(b1p + boff);

        acc00 = __builtin_amdgcn_wmma_f32_16x16x4_f32(false, af0, false, bf0,
                                                      (short)0, acc00, false, false);
        acc01 = __builtin_amdgcn_wmma_f32_16x16x4_f32(false, af0, false, bf1,
                                                      (short)0, acc01, false, false);
        acc10 = __builtin_amdgcn_wmma_f32_16x16x4_f32(false, af1, false, bf0,
                                                      (short)0, acc10, false, false);
        acc11 = __builtin_amdgcn_wmma_f32_16x16x4_f32(false, af1, false, bf1,
                                                      (short)0, acc11, false, false);
    }

    // C/D 16x16 layout: lane n = lane%16, VGPR v holds row M = v + 8*quad.
#pragma unroll
    for (int v = 0; v < 8; ++v) {
        const int row = v + 8 * quad;
        C[(size_t)(m0 + row)      * ldc + n0 + mn]      = acc00[v] + addC;
        C[(size_t)(m0 + row)      * ldc + n0 + 16 + mn] = acc01[v] + addC;
        C[(size_t)(m0 + 16 + row) * ldc + n0 + mn]      = acc10[v] + addC;
        C[(size_t)(m0 + 16 + row) * ldc + n0 + 16 + mn] = acc11[v] + addC;
    }
}

// ---------------------------------------------------------------------------
// Column-normalize D and emit BOTH packed dictionary operands:
//   DnPk : packed form of Dn  [N rows, K cols]  (B operand of du GEMM)
//   DnTPk: packed form of DnT [K rows, N cols]  (B operand of rec GEMMs)
// One block per column k.
// ---------------------------------------------------------------------------
__global__ void lca_normalize_cols(const float* __restrict__ D,
                                   float* __restrict__ DnPk,
                                   float* __restrict__ DnTPk)
{
    __shared__ float s[256];
    const int k = blockIdx.x;
    const int t = threadIdx.x;
    float sum = 0.f;
    for (int n = t; n < LCA_N; n += 256) {
        float v = D[(size_t)n * LCA_K + k];
        sum += v * v;
    }
    s[t] = sum;
    __syncthreads();
    for (int off = 128; off > 0; off >>= 1) {
        if (t < off) s[t] += s[t + off];
        __syncthreads();
    }
    const float inv = 1.0f / fmaxf(sqrtf(s[0]), LCA_EPS);
    for (int n = t; n < LCA_N; n += 256) {
        float v = D[(size_t)n * LCA_K + k] * inv;
        // Dn  packed: row n, col k  -> ((n>>1)*K + k)*2 + (n&1)
        DnPk [((((size_t)(n >> 1)) * LCA_K + k) << 1) | (n & 1)] = v;
        // DnT packed: row k, col n  -> ((k>>1)*N + n)*2 + (k&1)
        DnTPk[((((size_t)(k >> 1)) * LCA_N + n) << 1) | (k & 1)] = v;
    }
}

// ---------------------------------------------------------------------------
// State init: u=a=0, lr=1/K, best_loss=inf, done=0/upd=1. Grid covers B*K.
// ---------------------------------------------------------------------------
__global__ void lca_init_state(float* __restrict__ u, float* __restrict__ a,
                               float* __restrict__ lr, float* __restrict__ best,
                               int* __restrict__ done_upd)
{
    const int i = blockIdx.x * 256 + threadIdx.x;
    u[i] = 0.f;
    a[i] = 0.f;
    if (i < LCA_B) { lr[i] = 1.0f / (float)LCA_K; best[i] = INFINITY; }
    if (i == 0) { done_upd[0] = 0; done_upd[1] = 1; }
}

// ---------------------------------------------------------------------------
// loss[b] = mean((x-rec)^2) + lambda*mean(|a|); also writes r = rec - x.
// One block per sample b.
// ---------------------------------------------------------------------------
__global__ void lca_loss_resid(const float* __restrict__ x,
                               const float* __restrict__ rec,
                               const float* __restrict__ a,
                               float* __restrict__ r,
                               float* __restrict__ loss)
{
    __shared__ float se_s[256];
    __shared__ float sa_s[256];
    const int b = blockIdx.x;
    const int t = threadIdx.x;
    const float* xp = x   + (size_t)b * LCA_N;
    const float* rp = rec + (size_t)b * LCA_N;
    float* ro       = r   + (size_t)b * LCA_N;
    float se = 0.f;
    for (int n = t; n < LCA_N; n += 256) {
        float d = xp[n] - rp[n];
        se += d * d;
        ro[n] = -d;  // rec - x
    }
    const float* ap = a + (size_t)b * LCA_K;
    float sa = 0.f;
    for (int k = t; k < LCA_K; k += 256) sa += fabsf(ap[k]);
    se_s[t] = se; sa_s[t] = sa;
    __syncthreads();
    for (int off = 128; off > 0; off >>= 1) {
        if (t < off) { se_s[t] += se_s[t + off]; sa_s[t] += sa_s[t + off]; }
        __syncthreads();
    }
    if (t == 0)
        loss[b] = se_s[0] * (1.0f / LCA_N) + LCA_LAMBD * sa_s[0] * (1.0f / LCA_K);
}

// ---------------------------------------------------------------------------
// done |= max_b(best-loss) < TOL; upd = !done; best = upd ? loss : best.
// Single block of 256 threads.
// ---------------------------------------------------------------------------
__global__ void lca_done_update(const float* __restrict__ loss,
                                float* __restrict__ best,
                                int* __restrict__ done_upd)
{
    __shared__ float s[256];
    __shared__ int updS;
    const int b = threadIdx.x;
    s[b] = best[b] - loss[b];
    __syncthreads();
    for (int off = 128; off > 0; off >>= 1) {
        if (b < off) s[b] = fmaxf(s[b], s[b + off]);
        __syncthreads();
    }
    if (b == 0) {
        int done = done_upd[0];
        if (s[0] < LCA_TOL) done = 1;
        done_upd[0] = done;
        done_upd[1] = done ? 0 : 1;
        updS = done ? 0 : 1;
    }
    __syncthreads();
    if (updS) best[b] = loss[b];
}

// ---------------------------------------------------------------------------
// Line-search candidates: a_c = relu(u - du * lr*fac), fac in {0.5, 2.0}.
// du already includes +lambda from the GEMM epilogue.
// ---------------------------------------------------------------------------
__global__ void lca_candidates(const float* __restrict__ u,
                               const float* __restrict__ du,
                               const float* __restrict__ lr,
                               float* __restrict__ ac0,
                               float* __restrict__ ac1)
{
    const int i = blockIdx.x * 256 + threadIdx.x;
    const int b = i >> 12;  // /4096
    const float l = lr[b];
    const float d = du[i];
    const float uu = u[i];
    ac0[i] = fmaxf(uu - d * (l * 0.5f), 0.f);
    ac1[i] = fmaxf(uu - d * (l * 2.0f), 0.f);
}

// ---------------------------------------------------------------------------
// loss_c[c,b] for both candidates; blockIdx = (b, c).
// ---------------------------------------------------------------------------
__global__ void lca_loss_c(const float* __restrict__ x,
                           const float* __restrict__ rec_c,
                           const float* __restrict__ a_c,
                           float* __restrict__ loss_c)
{
    __shared__ float se_s[256];
    __shared__ float sa_s[256];
    const int b = blockIdx.x;
    const int c = blockIdx.y;
    const int t = threadIdx.x;
    const float* xp = x + (size_t)b * LCA_N;
    const float* rp = rec_c + (size_t)c * LCA_BN + (size_t)b * LCA_N;
    const float* ap = a_c   + (size_t)c * LCA_BK + (size_t)b * LCA_K;
    float se = 0.f, sa = 0.f;
    for (int n = t; n < LCA_N; n += 256) {
        float d = xp[n] - rp[n];
        se += d * d;
    }
    for (int k = t; k < LCA_K; k += 256) sa += fabsf(ap[k]);
    se_s[t] = se; sa_s[t] = sa;
    __syncthreads();
    for (int off = 128; off > 0; off >>= 1) {
        if (t < off) { se_s[t] += se_s[t + off]; sa_s[t] += sa_s[t + off]; }
        __syncthreads();
    }
    if (t == 0)
        loss_c[c * LCA_B + b] =
            se_s[0] * (1.0f / LCA_N) + LCA_LAMBD * sa_s[0] * (1.0f / LCA_K);
}

// ---------------------------------------------------------------------------
// Pick winner, update u and a (using OLD lr). Frozen when upd==0.
// ---------------------------------------------------------------------------
__global__ void lca_select(float* __restrict__ u, float* __restrict__ a,
                           const float* __restrict__ du,
                           const float* __restrict__ lr,
                           const float* __restrict__ loss_c,
                           const int* __restrict__ done_upd)
{
    if (!done_upd[1]) return;  // uniform across grid
    const int i = blockIdx.x * 256 + threadIdx.x;
    const int b = i >> 12;
    const bool pick0 = loss_c[b] <= loss_c[LCA_B + b];  // argmin, first on tie
    const float fac = pick0 ? 0.5f : 2.0f;
    const float un = u[i] - du[i] * (lr[b] * fac);
    u[i] = un;
    a[i] = fmaxf(un, 0.f);
}

// Runs AFTER lca_select (stream order) so lr update does not disturb u update.
__global__ void lca_lr_update(float* __restrict__ lr,
                              const float* __restrict__ loss_c,
                              const int* __restrict__ done_upd)
{
    if (!done_upd[1]) return;
    const int b = threadIdx.x;
    const bool pick0 = loss_c[b] <= loss_c[LCA_B + b];
    lr[b] *= pick0 ? 0.5f : 2.0f;
}

__global__ void lca_copy(const float* __restrict__ src, float* __restrict__ dst)
{
    const int i = blockIdx.x * 256 + threadIdx.x;
    dst[i] = src[i];
}

// ---------------------------------------------------------------------------
extern "C" void kernel_launch(void* const* d_in, const int* in_sizes, int n_in,
                              void* d_out, int out_size, void* d_ws, size_t ws_size,
                              hipStream_t stream)
{
    (void)in_sizes; (void)n_in; (void)out_size; (void)ws_size;
    const float* x = (const float*)d_in[0];  // [256, 1024]
    const float* D = (const float*)d_in[1];  // [1024, 4096]
    float* out = (float*)d_out;              // a [256*4096] then rec [256*1024]

    // Workspace layout (floats).
    float* ws     = (float*)d_ws;
    float* DnPk   = ws;                              // 4194304 (packed Dn)
    float* DnTPk  = DnPk   + (size_t)LCA_N * LCA_K;  // 4194304 (packed DnT)
    float* u      = DnTPk  + (size_t)LCA_K * LCA_N;  // 1048576
    float* a      = u      + LCA_BK;                 // 1048576
    float* du     = a      + LCA_BK;                 // 1048576
    float* ac     = du     + LCA_BK;                 // 2 x 1048576 (batched)
    float* rec    = ac     + 2 * (size_t)LCA_BK;     // 262144
    float* r      = rec    + LCA_BN;                 // 262144
    float* rec_c  = r      + LCA_BN;                 // 2 x 262144 (batched)
    float* loss   = rec_c  + 2 * (size_t)LCA_BN;     // 256
    float* loss_c = loss   + LCA_B;                  // 512
    float* best   = loss_c + 2 * LCA_B;              // 256
    float* lr     = best   + LCA_B;                  // 256
    int*   done_upd = (int*)(lr + LCA_B);            // [done, upd]

    // Dictionary normalization + packing (once) + state init.
    lca_normalize_cols<<<LCA_K, 256, 0, stream>>>(D, DnPk, DnTPk);
    lca_init_state<<<LCA_BK / 256, 256, 0, stream>>>(u, a, lr, best, done_upd);

    // Tile grids: (M/32)*(N/32) waves, 8 waves per block.
    const int gRec = (LCA_B / 32) * (LCA_N / 32) / 8;  // 32 blocks
    const int gDu  = (LCA_B / 32) * (LCA_K / 32) / 8;  // 128 blocks

    for (int it = 0; it < LCA_ITERS; ++it) {
        // rec = a @ Dn^T   (M=256, N=1024, K=4096)
        lca_wmma_gemm32<<<dim3(gRec, 1), 256, 0, stream>>>(
            a, DnTPk, rec, LCA_N, LCA_K, LCA_K, LCA_N, LCA_N, 0, 0, 0.f);
        lca_loss_resid<<<LCA_B, 256, 0, stream>>>(x, rec, a, r, loss);
        lca_done_update<<<1, 256, 0, stream>>>(loss, best, done_upd);
        // du = (rec - x) @ Dn + lambda   (M=256, N=4096, K=1024)
        lca_wmma_gemm32<<<dim3(gDu, 1), 256, 0, stream>>>(
            r, DnPk, du, LCA_K, LCA_N, LCA_N, LCA_K, LCA_K, 0, 0, LCA_LAMBD);
        lca_candidates<<<LCA_BK / 256, 256, 0, stream>>>(u, du, lr, ac, ac + LCA_BK);
        // rec_c[c] = a_c[c] @ Dn^T, batched over candidate c via gridDim.y
        lca_wmma_gemm32<<<dim3(gRec, 2), 256, 0, stream>>>(
            ac, DnTPk, rec_c, LCA_N, LCA_K, LCA_K, LCA_N, LCA_N,
            (long)LCA_BK, (long)LCA_BN, 0.f);
        lca_loss_c<<<dim3(LCA_B, 2), 256, 0, stream>>>(x, rec_c, ac, loss_c);
        lca_select<<<LCA_BK / 256, 256, 0, stream>>>(u, a, du, lr, loss_c, done_upd);
        lca_lr_update<<<1, 256, 0, stream>>>(lr, loss_c, done_upd);
    }

    // Outputs: a, then rec = a @ Dn^T with final a.
    lca_copy<<<LCA_BK / 256, 256, 0, stream>>>(a, out);
    lca_wmma_gemm32<<<dim3(gRec, 1), 256, 0, stream>>>(
        a, DnTPk, out + LCA_BK, LCA_N, LCA_K, LCA_K, LCA_N, LCA_N, 0, 0, 0.f);
}